// Criterion_64965675319884
// MI455X (gfx1250) — compile-verified
//
#include <hip/hip_runtime.h>
#include <math.h>

#define BN 384
#define DM 512
#define G  4
#define TAUF 3.0f
#define ANNEAL_INV 100.0f

typedef float v2f __attribute__((ext_vector_type(2)));
typedef float v8f __attribute__((ext_vector_type(8)));

// ---------------------------------------------------------------------------
// Kernel 1: fp32 Gram matrices via V_WMMA_F32_16X16X4_F32.
// One wave computes one 16x16 tile of S = X X^T (grid.z selects student/teacher).
// A fragment (16x4):  lanes 0-15 hold M=lane, K=0..1 ; lanes 16-31 hold M=lane-16, K=2..3
// B fragment (4x16):  identical pattern with N in place of M (symmetric product),
// so both fragments are plain float2 loads from rows of X.
// ---------------------------------------------------------------------------
__global__ __launch_bounds__(32)
void gram_wmma(const float* __restrict__ X0, const float* __restrict__ X1,
               float* __restrict__ S0, float* __restrict__ S1)
{
    const float* X = blockIdx.z ? X1 : X0;
    float*       S = blockIdx.z ? S1 : S0;
    const int i0   = blockIdx.y * 16;
    const int j0   = blockIdx.x * 16;
    const int lane = threadIdx.x & 31;
    const int row  = lane & 15;
    const int kh   = (lane >> 4) << 1;            // 0 or 2
    const float* arow = X + (size_t)(i0 + row) * DM + kh;
    const float* brow = X + (size_t)(j0 + row) * DM + kh;

    v8f acc = {0.f, 0.f, 0.f, 0.f, 0.f, 0.f, 0.f, 0.f};
#pragma unroll 8
    for (int d = 0; d < DM; d += 4) {
        v2f a = *(const v2f*)(arow + d);
        v2f b = *(const v2f*)(brow + d);
        acc = __builtin_amdgcn_wmma_f32_16x16x4_f32(
            /*neg_a=*/false, a, /*neg_b=*/false, b,
            /*c_mod=*/(short)0, acc, /*reuse_a=*/false, /*reuse_b=*/false);
    }
    // C/D layout: VGPR v -> lanes 0-15: (M=v, N=lane); lanes 16-31: (M=v+8, N=lane-16)
    const int n  = lane & 15;
    const int mb = (lane >> 4) << 3;
#pragma unroll
    for (int v = 0; v < 8; ++v)
        S[(size_t)(i0 + mb + v) * BN + (j0 + n)] = acc[v];
}

// ---------------------------------------------------------------------------
// helpers
// ---------------------------------------------------------------------------
__device__ __forceinline__ float sig_clip(float d)
{
    float t = fminf(fmaxf(d * ANNEAL_INV, -50.f), 50.f);
    return 1.0f / (1.0f + __expf(-t));
}

__device__ __forceinline__ float block_sum(float v, float* red, int tid)
{
    red[tid] = v;
    __syncthreads();
    for (int s = 64; s > 0; s >>= 1) {
        if (tid < s) red[tid] += red[tid + s];
        __syncthreads();
    }
    float r = red[0];
    __syncthreads();
    return r;
}

__device__ __forceinline__ float block_max(float v, float* red, int tid)
{
    red[tid] = v;
    __syncthreads();
    for (int s = 64; s > 0; s >>= 1) {
        if (tid < s) red[tid] = fmaxf(red[tid], red[tid + s]);
        __syncthreads();
    }
    float r = red[0];
    __syncthreads();
    return r;
}

// ---------------------------------------------------------------------------
// Kernel 2: rank term. Only the diagonal blocks of rk_all are consumed, so per
// row i we need 4 sums over k of sigmoid((sim[i,k]-sim[i,blk+jj])/0.01), plus
// the 4x4 rk_pos built from the same diagonal block of sim.
// ---------------------------------------------------------------------------
__global__ __launch_bounds__(128)
void rank_kernel(const float* __restrict__ sim, float* __restrict__ rank_row)
{
    __shared__ float red[128];
    const int i   = blockIdx.x;
    const int tid = threadIdx.x;
    const int blk = i & ~(G - 1);
    const float* rowp = sim + (size_t)i * BN;

    float sj[G];
#pragma unroll
    for (int jj = 0; jj < G; ++jj) sj[jj] = rowp[blk + jj];

    float part[G] = {0.f, 0.f, 0.f, 0.f};
    for (int k = tid; k < BN; k += 128) {
        float sik = rowp[k];
#pragma unroll
        for (int jj = 0; jj < G; ++jj)
            if (k != blk + jj)                       // mask removes k == j only
                part[jj] += sig_clip(sik - sj[jj]);
    }
    float rk[G];
#pragma unroll
    for (int jj = 0; jj < G; ++jj)
        rk[jj] = block_sum(part[jj], red, tid) + 1.0f;

    if (tid == 0) {
        float contrib = 0.f;
#pragma unroll
        for (int jj = 0; jj < G; ++jj) {
            float rp = 1.0f;                         // rk_pos[n, i%4, jj]
#pragma unroll
            for (int kk = 0; kk < G; ++kk)
                if (kk != jj) rp += sig_clip(sj[kk] - sj[jj]);
            contrib += rp / rk[jj];
        }
        rank_row[i] = contrib;
    }
}

// ---------------------------------------------------------------------------
// Kernel 3: KD term. Per row i: stable log-softmax of scaled student/teacher
// similarities, KL(q || p). 384 cols / 128 threads = 3 per thread (registers).
// ---------------------------------------------------------------------------
__global__ __launch_bounds__(128)
void kd_kernel(const float* __restrict__ sim, const float* __restrict__ tsim,
               const int* __restrict__ labels, float* __restrict__ kd_row)
{
    __shared__ float red[128];
    const int i   = blockIdx.x;
    const int tid = threadIdx.x;
    const int li  = labels[i];

    float av[3], bv[3];
    float ma = -1e30f, mb = -1e30f;
#pragma unroll
    for (int c = 0; c < 3; ++c) {
        int j = tid + c * 128;
        float sc = (labels[j] == li) ? 1.0f : 0.5f;
        float a  = sim [(size_t)i * BN + j] * sc * (1.0f / TAUF);
        float b  = tsim[(size_t)i * BN + j] * sc * (1.0f / TAUF);
        av[c] = a; bv[c] = b;
        ma = fmaxf(ma, a); mb = fmaxf(mb, b);
    }
    ma = block_max(ma, red, tid);
    mb = block_max(mb, red, tid);

    float sa = 0.f, sb = 0.f;
#pragma unroll
    for (int c = 0; c < 3; ++c) {
        sa += __expf(av[c] - ma);
        sb += __expf(bv[c] - mb);
    }
    sa = block_sum(sa, red, tid);
    sb = block_sum(sb, red, tid);
    const float lsea = ma + __logf(sa);
    const float lseb = mb + __logf(sb);

    float kl = 0.f;
#pragma unroll
    for (int c = 0; c < 3; ++c) {
        float lq = bv[c] - lseb;                     // log q
        kl += __expf(lq) * (lq - (av[c] - lsea));    // q * (log q - log p)
    }
    kl = block_sum(kl, red, tid);
    if (tid == 0) kd_row[i] = kl;
}

// ---------------------------------------------------------------------------
// Kernel 4: combine.
// ---------------------------------------------------------------------------
__global__ __launch_bounds__(128)
void finalize_kernel(const float* __restrict__ rank_row,
                     const float* __restrict__ kd_row,
                     const int* __restrict__ epoch,
                     float* __restrict__ out)
{
    __shared__ float red[128];
    const int tid = threadIdx.x;
    float rs = 0.f, ks = 0.f;
    for (int j = tid; j < BN; j += 128) { rs += rank_row[j]; ks += kd_row[j]; }
    rs = block_sum(rs, red, tid);
    ks = block_sum(ks, red, tid);
    if (tid == 0) {
        float loss_rank = 1.0f - rs / (float)(G * BN);
        float loss_kd   = ks / (float)BN;
        float e         = (float)(*epoch);
        float loss      = loss_rank + (e / 100.0f) * (TAUF * TAUF) * loss_kd;
        out[0] = loss;
        out[1] = loss_rank;
        out[2] = loss_kd;
    }
}

// ---------------------------------------------------------------------------
extern "C" void kernel_launch(void* const* d_in, const int* in_sizes, int n_in,
                              void* d_out, int out_size, void* d_ws, size_t ws_size,
                              hipStream_t stream)
{
    (void)in_sizes; (void)n_in; (void)out_size; (void)ws_size;
    const float* batch   = (const float*)d_in[0];
    const float* teacher = (const float*)d_in[1];
    const int*   labels  = (const int*)d_in[2];
    const int*   epoch   = (const int*)d_in[3];

    float* ws       = (float*)d_ws;
    float* sim      = ws;                         // 384*384
    float* tsim     = sim + (size_t)BN * BN;      // 384*384
    float* rank_row = tsim + (size_t)BN * BN;     // 384
    float* kd_row   = rank_row + BN;              // 384

    dim3 ggrid(BN / 16, BN / 16, 2);
    gram_wmma<<<ggrid, 32, 0, stream>>>(batch, teacher, sim, tsim);
    rank_kernel<<<BN, 128, 0, stream>>>(sim, rank_row);
    kd_kernel<<<BN, 128, 0, stream>>>(sim, tsim, labels, kd_row);
    finalize_kernel<<<1, 128, 0, stream>>>(rank_row, kd_row, epoch, (float*)d_out);
}